// MAB_9363028705889
// MI455X (gfx1250) — compile-verified
//
#include <hip/hip_runtime.h>
#include <hip/hip_bf16.h>

typedef __attribute__((ext_vector_type(2))) float v2f;
typedef __attribute__((ext_vector_type(8))) float v8f;

#define R_ROWS 49152          // B*T*N = 8*12*512
#define NELEM  (R_ROWS * 64)  // total activation elements
#define EPS    1e-5f

// ---------------------------------------------------------------------------
// gemm64: Y[r, c] = sum_k X[r,k] * W[k,c] + bias[c]   (X: [R,64], W: [64,64])
// One wave computes a 16x64 output stripe with V_WMMA_F32_16X16X4_F32.
// A layout: lane<16 -> row=lane, {K0,K1}; lane>=16 -> row=lane-16, {K2,K3}
// B layout (mirror): lane<16 -> col=lane, {K0,K1}; lane>=16 -> {K2,K3}
// D layout: VGPR j, lane l -> row j + 8*(l>=16), col (l&15)
// ---------------------------------------------------------------------------
__global__ __launch_bounds__(128) void gemm64_kernel(
    const float* __restrict__ X, const float* __restrict__ W,
    const float* __restrict__ bias, float* __restrict__ Y)
{
  const int lane = threadIdx.x & 31;
  const int wid  = threadIdx.x >> 5;
  const int hi   = (lane >> 4) & 1;
  const int lo   = lane & 15;
  const int r0   = blockIdx.x * 64 + wid * 16;

  v8f acc[4] = {{}, {}, {}, {}};
#pragma unroll
  for (int k0 = 0; k0 < 64; k0 += 4) {
    v2f a;
    const float* xp = X + (size_t)(r0 + lo) * 64 + (k0 + 2 * hi);
    a.x = xp[0];
    a.y = xp[1];
#pragma unroll
    for (int ct = 0; ct < 4; ++ct) {
      v2f bm;
      const float* wp = W + (size_t)(k0 + 2 * hi) * 64 + ct * 16 + lo;
      bm.x = wp[0];
      bm.y = wp[64];
      acc[ct] = __builtin_amdgcn_wmma_f32_16x16x4_f32(
          false, a, false, bm, (short)0, acc[ct], false, false);
    }
  }
#pragma unroll
  for (int ct = 0; ct < 4; ++ct) {
    const int c = ct * 16 + lo;
    const float bv = bias[c];
#pragma unroll
    for (int jj = 0; jj < 8; ++jj) {
      Y[(size_t)(r0 + jj + 8 * hi) * 64 + c] = acc[ct][jj] + bv;
    }
  }
}

// ---------------------------------------------------------------------------
// Deterministic per-channel sum / sum-of-squares (64 blocks, one per channel)
// ---------------------------------------------------------------------------
__global__ __launch_bounds__(256) void stats_kernel(
    const float* __restrict__ Y, float* __restrict__ stats)
{
  __shared__ float ssum[256];
  __shared__ float ssq[256];
  const int c = blockIdx.x;
  float s = 0.0f, q = 0.0f;
  for (int r = threadIdx.x; r < R_ROWS; r += 256) {
    float v = Y[(size_t)r * 64 + c];
    s += v;
    q += v * v;
  }
  ssum[threadIdx.x] = s;
  ssq[threadIdx.x]  = q;
  __syncthreads();
  for (int off = 128; off > 0; off >>= 1) {
    if (threadIdx.x < off) {
      ssum[threadIdx.x] += ssum[threadIdx.x + off];
      ssq[threadIdx.x]  += ssq[threadIdx.x + off];
    }
    __syncthreads();
  }
  if (threadIdx.x == 0) {
    stats[c]      = ssum[0];
    stats[64 + c] = ssq[0];
  }
}

// BatchNorm (train-mode batch stats, biased var) + ReLU, elementwise.
__global__ void bn_apply_kernel(const float* __restrict__ Y,
                                const float* __restrict__ stats,
                                const float* __restrict__ g,
                                const float* __restrict__ beta,
                                float* __restrict__ dst, int n)
{
  int i = blockIdx.x * blockDim.x + threadIdx.x;
  if (i >= n) return;
  const int c = i & 63;
  const float invR = 1.0f / (float)R_ROWS;
  float mu  = stats[c] * invR;
  float var = stats[64 + c] * invR - mu * mu;
  float x   = Y[i];
  float y   = g[c] * (x - mu) * rsqrtf(var + EPS) + beta[c];
  dst[i] = fmaxf(y, 0.0f);
}

// mrow[hb, t, n] = sum over the 8 mask channels of head h (hb = h*8 + b)
__global__ void mrow_kernel(const float* __restrict__ mask,
                            float* __restrict__ mrow)
{
  int i = blockIdx.x * blockDim.x + threadIdx.x;
  if (i >= 64 * 12 * 512) return;
  const int n  = i & 511;
  const int t  = (i >> 9) % 12;
  const int hb = i / (12 * 512);
  const int h  = hb >> 3;
  const int b  = hb & 7;
  const float* mp = mask + ((size_t)(b * 12 + t) * 512 + n) * 64 + h * 8;
  float s = 0.0f;
#pragma unroll
  for (int c = 0; c < 8; ++c) s += mp[c];
  mrow[i] = s;
}

// ---------------------------------------------------------------------------
// Flash attention: one block per (qblock, t, hb); 4 waves, 16 query rows each.
// scores tile 16x16 via 2 f32 WMMAs (K=8), streaming softmax with lane
// butterflies (xor 1/2/4/8 stays within each 16-lane half = one C-layout row
// group), then O += P@V via 4 f32 WMMAs with P staged through LDS.
// ---------------------------------------------------------------------------
__global__ __launch_bounds__(128) void attn_kernel(
    const float* __restrict__ q, const float* __restrict__ k,
    const float* __restrict__ v, const float* __restrict__ mrow,
    float* __restrict__ out)
{
  __shared__ float lds[4][16][16];
  const int lane = threadIdx.x & 31;
  const int wid  = threadIdx.x >> 5;
  const int hi   = (lane >> 4) & 1;
  const int lo   = lane & 15;

  const int qb = blockIdx.x;   // 0..7
  const int t  = blockIdx.y;   // 0..11
  const int hb = blockIdx.z;   // 0..63 (h*8 + b)
  const int h  = hb >> 3;
  const int b  = hb & 7;

  const int n0 = qb * 64 + wid * 16;
  const size_t base = (size_t)(b * 12 + t) * 512 * 64;
  const int ch0 = h * 8;
  const float scale = 0.35355339059327376f;  // 1/sqrt(8)

  // Query A-tile (16x8 -> two K=4 steps)
  v2f aq[2];
  {
    const float* qp = q + base + (size_t)(n0 + lo) * 64 + ch0;
    aq[0].x = qp[0 + 2 * hi];
    aq[0].y = qp[1 + 2 * hi];
    aq[1].x = qp[4 + 2 * hi];
    aq[1].y = qp[5 + 2 * hi];
  }
  // Per-row mask flags for the 8 rows of this lane's accumulator slots
  float mr[8];
#pragma unroll
  for (int jj = 0; jj < 8; ++jj)
    mr[jj] = mrow[((size_t)hb * 12 + t) * 512 + (n0 + jj + 8 * hi)];

  float m_run[8], l_run[8];
#pragma unroll
  for (int jj = 0; jj < 8; ++jj) { m_run[jj] = -3.0e38f; l_run[jj] = 0.0f; }
  v8f o = {};

  for (int j0 = 0; j0 < 512; j0 += 16) {
    // K^T B-tile
    v2f bk0, bk1;
    {
      const float* kp = k + base + (size_t)(j0 + lo) * 64 + ch0;
      bk0.x = kp[0 + 2 * hi];
      bk0.y = kp[1 + 2 * hi];
      bk1.x = kp[4 + 2 * hi];
      bk1.y = kp[5 + 2 * hi];
    }
    v8f c = {};
    c = __builtin_amdgcn_wmma_f32_16x16x4_f32(false, aq[0], false, bk0,
                                              (short)0, c, false, false);
    c = __builtin_amdgcn_wmma_f32_16x16x4_f32(false, aq[1], false, bk1,
                                              (short)0, c, false, false);

    float p[8];
#pragma unroll
    for (int jj = 0; jj < 8; ++jj) {
      float sc = c[jj] * scale;
      sc = (mr[jj] == 0.0f) ? -1.0e9f : sc;
      float tm = sc;
      tm = fmaxf(tm, __shfl_xor(tm, 1, 32));
      tm = fmaxf(tm, __shfl_xor(tm, 2, 32));
      tm = fmaxf(tm, __shfl_xor(tm, 4, 32));
      tm = fmaxf(tm, __shfl_xor(tm, 8, 32));
      float mnew = fmaxf(m_run[jj], tm);
      float pe = __expf(sc - mnew);
      float ps = pe;
      ps += __shfl_xor(ps, 1, 32);
      ps += __shfl_xor(ps, 2, 32);
      ps += __shfl_xor(ps, 4, 32);
      ps += __shfl_xor(ps, 8, 32);
      float alpha = __expf(m_run[jj] - mnew);
      l_run[jj] = l_run[jj] * alpha + ps;
      m_run[jj] = mnew;
      o[jj] *= alpha;
      p[jj] = pe;
    }
    // Stage P (C layout) -> LDS (row-major) for A-layout reload
#pragma unroll
    for (int jj = 0; jj < 8; ++jj)
      lds[wid][jj + 8 * hi][lo] = p[jj];
    __syncthreads();
    // O += P(16x16) @ V(16x8, zero-padded to 16 cols): 4 K=4 steps
#pragma unroll
    for (int s = 0; s < 4; ++s) {
      v2f ap, bv;
      ap.x = lds[wid][lo][4 * s + 2 * hi];
      ap.y = lds[wid][lo][4 * s + 2 * hi + 1];
      const int kk = j0 + 4 * s + 2 * hi;
      const float* vp = v + base + (size_t)kk * 64 + ch0 + lo;
      bv.x = (lo < 8) ? vp[0]  : 0.0f;
      bv.y = (lo < 8) ? vp[64] : 0.0f;
      o = __builtin_amdgcn_wmma_f32_16x16x4_f32(false, ap, false, bv,
                                                (short)0, o, false, false);
    }
    __syncthreads();
  }
  // Normalize and scatter back in merged-heads layout [B,T,N,D]
  if (lo < 8) {
#pragma unroll
    for (int jj = 0; jj < 8; ++jj) {
      float val = o[jj] / l_run[jj];
      out[base + (size_t)(n0 + jj + 8 * hi) * 64 + ch0 + lo] = val;
    }
  }
}

// ---------------------------------------------------------------------------
extern "C" void kernel_launch(void* const* d_in, const int* in_sizes, int n_in,
                              void* d_out, int out_size, void* d_ws,
                              size_t ws_size, hipStream_t stream)
{
  const float* Q     = (const float*)d_in[0];
  const float* Kin   = (const float*)d_in[1];
  const float* mask  = (const float*)d_in[2];
  const float* Wq    = (const float*)d_in[4];
  const float* bq    = (const float*)d_in[5];
  const float* gq    = (const float*)d_in[6];
  const float* betaq = (const float*)d_in[7];
  const float* Wk    = (const float*)d_in[8];
  const float* bk    = (const float*)d_in[9];
  const float* gk    = (const float*)d_in[10];
  const float* betak = (const float*)d_in[11];
  const float* Wv    = (const float*)d_in[12];
  const float* bv    = (const float*)d_in[13];
  const float* gv    = (const float*)d_in[14];
  const float* betav = (const float*)d_in[15];
  const float* Wo    = (const float*)d_in[16];
  const float* bo    = (const float*)d_in[17];
  const float* go    = (const float*)d_in[18];
  const float* betao = (const float*)d_in[19];

  float* ws = (float*)d_ws;
  const size_t SZ = (size_t)NELEM;           // 3,145,728 floats per tensor
  float* Yq = ws;                            // reused as Yo after attention
  float* Yk = ws + SZ;
  float* Yv = ws + 2 * SZ;
  float* Ao = ws + 3 * SZ;                   // merged attention output
  float* Mr = ws + 4 * SZ;                   // 64*12*512 row-mask sums
  float* st = Mr + (size_t)64 * 12 * 512;    // 4 x {sum[64], sumsq[64]}
  float* stQ = st, *stK = st + 128, *stV = st + 256, *stO = st + 384;

  const dim3 gb(R_ROWS / 64), tb(128);

  // Projections (GEMM + bias), deterministic BN stats, BN+ReLU in place
  gemm64_kernel<<<gb, tb, 0, stream>>>(Q, Wq, bq, Yq);
  gemm64_kernel<<<gb, tb, 0, stream>>>(Kin, Wk, bk, Yk);
  gemm64_kernel<<<gb, tb, 0, stream>>>(Kin, Wv, bv, Yv);
  stats_kernel<<<64, 256, 0, stream>>>(Yq, stQ);
  stats_kernel<<<64, 256, 0, stream>>>(Yk, stK);
  stats_kernel<<<64, 256, 0, stream>>>(Yv, stV);
  const int nbe = (NELEM + 255) / 256;
  bn_apply_kernel<<<nbe, 256, 0, stream>>>(Yq, stQ, gq, betaq, Yq, NELEM);
  bn_apply_kernel<<<nbe, 256, 0, stream>>>(Yk, stK, gk, betak, Yk, NELEM);
  bn_apply_kernel<<<nbe, 256, 0, stream>>>(Yv, stV, gv, betav, Yv, NELEM);

  // Row-mask sums, then fused flash attention (no score materialization)
  mrow_kernel<<<(64 * 12 * 512 + 255) / 256, 256, 0, stream>>>(mask, Mr);
  attn_kernel<<<dim3(8, 12, 64), 128, 0, stream>>>(Yq, Yk, Yv, Mr, Ao);

  // Output projection + BN + ReLU -> d_out (Yq reused as Yo scratch)
  gemm64_kernel<<<gb, tb, 0, stream>>>(Ao, Wo, bo, Yq);
  stats_kernel<<<64, 256, 0, stream>>>(Yq, stO);
  bn_apply_kernel<<<nbe, 256, 0, stream>>>(Yq, stO, go, betao, (float*)d_out,
                                           NELEM);
}